// GNNLayer_56942676411149
// MI455X (gfx1250) — compile-verified
//
#include <hip/hip_runtime.h>
#include <hip/hip_bf16.h>
#include <math.h>

typedef __attribute__((ext_vector_type(2))) float v2f;
typedef __attribute__((ext_vector_type(8))) float v8f;

// ---------------------------------------------------------------------------
// C[M x 128] = A[M x K] @ W[128 x K]^T (+ bias[128]),   K compile-time.
// One block = 256 threads = 8 waves; block owns one 16-row M tile, wave w
// owns N-columns [16w,16w+16). A staged through LDS (stride 260 words:
// 260 mod 64 = 4 -> each lane's b64 fragment read hits a disjoint bank quad).
// V_WMMA_F32_16X16X4_F32: full-fp32 matrix pipe, K-loop unrolled x8.
// ---------------------------------------------------------------------------
template <int K>
__global__ __launch_bounds__(256) void gemm_nt128_wmma(
    const float* __restrict__ A, const float* __restrict__ W,
    const float* __restrict__ bias, float* __restrict__ C,
    int M)
{
    constexpr int KC     = (K > 256) ? 256 : K;   // K chunk staged in LDS
    constexpr int LDSTR  = 260;                   // padded row stride (words)
    __shared__ float lds[16 * LDSTR];

    const int tid   = threadIdx.x;
    const int mbase = blockIdx.x * 16;

    const int lane = tid & 31;
    const int wv   = tid >> 5;                    // 0..7 -> N tile
    const int nl   = lane & 15;                   // N in tile / A-frag row
    const int kh   = lane >> 4;                   // K-pair select {0,1}/{2,3}
    const int n    = wv * 16 + nl;                // output column (N = 128)

    const float bn = bias ? bias[n] : 0.0f;
    v8f acc;
    #pragma unroll
    for (int v = 0; v < 8; ++v) acc[v] = bn;

    const float* __restrict__ wrow = W + (size_t)n * K;    // W row n (N x K)
    const float* __restrict__ arow = lds + nl * LDSTR;     // A tile row m=nl

    for (int kc = 0; kc < K; kc += KC) {
        // cooperative stage of A[mbase:+16, kc:+KC] (zero-pad M overhang)
        #pragma unroll
        for (int i = 0; i < (16 * KC) / 256; ++i) {
            const int idx = tid + i * 256;
            const int r = idx / KC, c = idx % KC;          // compile-time KC
            const int m = mbase + r;
            lds[r * LDSTR + c] = (m < M) ? A[(size_t)m * K + kc + c] : 0.0f;
        }
        __syncthreads();

        #pragma unroll 8                                    // 8 WMMA in flight
        for (int k0 = 0; k0 < KC; k0 += 4) {
            const int k = k0 + 2 * kh;
            v2f a, b;
            a.x = arow[k];        a.y = arow[k + 1];        // A: M=nl, K=k,k+1
            b.x = wrow[kc + k];   b.y = wrow[kc + k + 1];   // B[k][n] = W[n][k]
            acc = __builtin_amdgcn_wmma_f32_16x16x4_f32(
                      false, a, false, b, (short)0, acc, false, false);
        }
        __syncthreads();
    }

    if (mbase + 16 <= M) {                    // fast path: full tile, no guards
        #pragma unroll
        for (int v = 0; v < 8; ++v)
            C[(size_t)(mbase + v + 8 * kh) * 128 + n] = acc[v];
    } else {
        #pragma unroll
        for (int v = 0; v < 8; ++v) {
            const int m = mbase + v + 8 * kh; // C/D layout: M = v + 8*(lane/16)
            if (m < M) C[(size_t)m * 128 + n] = acc[v];
        }
    }
}

// ---------------------------------------------------------------------------
// hqW[q][d] = sum_k hr_all[q_rel[q]][k] * Wqr_w[d][k] + Wqr_b[d]   (64 x 128)
// ---------------------------------------------------------------------------
__global__ void hq_kernel(const int* __restrict__ q_rel,
                          const float* __restrict__ hr_all,
                          const float* __restrict__ Wqr_w,
                          const float* __restrict__ Wqr_b,
                          float* __restrict__ hqW)
{
    const int q = blockIdx.x;
    const int d = threadIdx.x;               // 128 threads
    const int r = q_rel[q];
    const float* __restrict__ a = hr_all + (size_t)r * 128;
    const float* __restrict__ w = Wqr_w + (size_t)d * 128;
    float s = Wqr_b[d];
    #pragma unroll 4
    for (int k = 0; k < 128; ++k) s += a[k] * w[k];
    hqW[q * 128 + d] = s;
}

__global__ void zero_kernel(float* __restrict__ p, size_t n)
{
    size_t i = (size_t)blockIdx.x * blockDim.x + threadIdx.x;
    if (i < n) p[i] = 0.0f;
}

// ---------------------------------------------------------------------------
// One wave (32 lanes) per edge; lane owns 4 contiguous dims (float4).
// Edge id forced wave-uniform via readfirstlane -> scalar index loads and
// SADDR-form coalesced gathers. Scatter uses non-returning f32 atomics.
// ---------------------------------------------------------------------------
__global__ __launch_bounds__(256) void edge_kernel(
    const int* __restrict__ edges, const int* __restrict__ r_idx,
    const float* __restrict__ hsW, const float* __restrict__ hrW,
    const float* __restrict__ hqW,
    const float* __restrict__ hidden, const float* __restrict__ hr_all,
    const float* __restrict__ walpha_w, const float* __restrict__ walpha_b,
    float* __restrict__ agg, int n_edge)
{
    const int e = __builtin_amdgcn_readfirstlane(
                      (int)((blockIdx.x * 256 + threadIdx.x) >> 5));
    const int lane = threadIdx.x & 31;
    if (e >= n_edge) return;                 // whole wave exits together

    const int sub = edges[3 * e + 0];
    const int rel = edges[3 * e + 1];
    const int obj = edges[3 * e + 2];
    const int rq  = r_idx[e];

    const float4* __restrict__ hsW4 = (const float4*)hsW;
    const float4* __restrict__ hrW4 = (const float4*)hrW;
    const float4* __restrict__ hqW4 = (const float4*)hqW;
    const float4* __restrict__ hid4 = (const float4*)hidden;
    const float4* __restrict__ hra4 = (const float4*)hr_all;
    const float4* __restrict__ wa4  = (const float4*)walpha_w;

    const float4 s4 = hsW4[(size_t)sub * 32 + lane];
    const float4 r4 = hrW4[(size_t)rel * 32 + lane];
    const float4 q4 = hqW4[(size_t)rq  * 32 + lane];

    const float tx = fmaxf(s4.x + r4.x + q4.x, 0.0f);
    const float ty = fmaxf(s4.y + r4.y + q4.y, 0.0f);
    const float tz = fmaxf(s4.z + r4.z + q4.z, 0.0f);
    const float tw = fmaxf(s4.w + r4.w + q4.w, 0.0f);

    const float4 w4 = wa4[lane];
    float p = tx * w4.x + ty * w4.y + tz * w4.z + tw * w4.w;
    #pragma unroll
    for (int off = 16; off > 0; off >>= 1) p += __shfl_xor(p, off, 32);

    const float alpha = 1.0f / (1.0f + expf(-(p + walpha_b[0])));

    const float4 h4 = hid4[(size_t)sub * 32 + lane];
    const float4 e4 = hra4[(size_t)rel * 32 + lane];
    float* dst = agg + (size_t)obj * 128 + lane * 4;
    unsafeAtomicAdd(dst + 0, alpha * h4.x * e4.x);
    unsafeAtomicAdd(dst + 1, alpha * h4.y * e4.y);
    unsafeAtomicAdd(dst + 2, alpha * h4.z * e4.z);
    unsafeAtomicAdd(dst + 3, alpha * h4.w * e4.w);
}

// ---------------------------------------------------------------------------
extern "C" void kernel_launch(void* const* d_in, const int* in_sizes, int n_in,
                              void* d_out, int out_size, void* d_ws, size_t ws_size,
                              hipStream_t stream)
{
    // inputs (setup_inputs order); d_in[0] = q_sub is unused by the reference
    const int*   q_rel    = (const int*)  d_in[1];
    const int*   r_idx    = (const int*)  d_in[2];
    const float* hidden   = (const float*)d_in[3];
    const int*   edges    = (const int*)  d_in[4];
    const float* Ws       = (const float*)d_in[6];
    const float* Wr       = (const float*)d_in[7];
    const float* Wqr_w    = (const float*)d_in[8];
    const float* Wqr_b    = (const float*)d_in[9];
    const float* walpha_w = (const float*)d_in[10];
    const float* walpha_b = (const float*)d_in[11];
    const float* Wh       = (const float*)d_in[12];
    const float* rela     = (const float*)d_in[13];
    const float* proj_w   = (const float*)d_in[14];
    const float* proj_b   = (const float*)d_in[15];

    const int DIM = 128, LLM = 1024;
    const int n_node = in_sizes[3]  / DIM;    // 50000
    const int n_edge = in_sizes[2];           // 500000
    const int n_rel  = in_sizes[13] / LLM;    // 500
    const int n_q    = in_sizes[1];           // 64

    // workspace carve-out (256B aligned): ~51.8 MB total
    char* ws = (char*)d_ws;
    auto carve = [&](size_t bytes) {
        char* p = ws; ws += (bytes + 255) & ~(size_t)255; return p;
    };
    float* hr_all = (float*)carve((size_t)n_rel  * DIM * sizeof(float));
    float* hrW    = (float*)carve((size_t)n_rel  * DIM * sizeof(float));
    float* hqW    = (float*)carve((size_t)n_q    * DIM * sizeof(float));
    float* hsW    = (float*)carve((size_t)n_node * DIM * sizeof(float));
    float* agg    = (float*)carve((size_t)n_node * DIM * sizeof(float));

    // 1) agg = 0
    const size_t tot = (size_t)n_node * DIM;
    zero_kernel<<<(unsigned)((tot + 255) / 256), 256, 0, stream>>>(agg, tot);

    // 2) hr_all = rela_embed @ proj_w^T + proj_b        (M=n_rel, K=1024)
    gemm_nt128_wmma<1024><<<(n_rel + 15) / 16, 256, 0, stream>>>(
        rela, proj_w, proj_b, hr_all, n_rel);

    // 3) hrW = hr_all @ Wr^T                            (M=n_rel, K=128)
    gemm_nt128_wmma<128><<<(n_rel + 15) / 16, 256, 0, stream>>>(
        hr_all, Wr, nullptr, hrW, n_rel);

    // 4) hqW = hr_all[q_rel] @ Wqr^T + Wqr_b            (64 x 128, tiny)
    hq_kernel<<<n_q, DIM, 0, stream>>>(q_rel, hr_all, Wqr_w, Wqr_b, hqW);

    // 5) hsW = hidden @ Ws^T                            (M=n_node, K=128)
    gemm_nt128_wmma<128><<<(n_node + 15) / 16, 256, 0, stream>>>(
        hidden, Ws, nullptr, hsW, n_node);

    // 6) per-edge attention + scatter-add into agg
    edge_kernel<<<(n_edge + 7) / 8, 256, 0, stream>>>(
        edges, r_idx, hsW, hrW, hqW, hidden, hr_all,
        walpha_w, walpha_b, agg, n_edge);

    // 7) out = agg @ Wh^T                               (M=n_node, K=128)
    gemm_nt128_wmma<128><<<(n_node + 15) / 16, 256, 0, stream>>>(
        agg, Wh, nullptr, (float*)d_out, n_node);
}